// DA3CrossFrameRKDAngleLoss_36524401885582
// MI455X (gfx1250) — compile-verified
//
#include <hip/hip_runtime.h>
#include <math.h>
#include <stdint.h>

typedef __attribute__((ext_vector_type(2))) float v2f;
typedef __attribute__((ext_vector_type(8))) float v8f;

#define B_  4
#define P_  1024
#define D_  1024
#define R_  256
#define S_  256
#define K_  4
#define E_  4096
#define RK_ 1024

// ---------------------------------------------------------------------------
// CDNA5 async global->LDS copy path (ASYNCcnt), guarded so the sync float4
// path is used if this toolchain lacks the builtins.
// Probe-learned prototype (from hipcc diagnostic):
//   __builtin_amdgcn_global_load_async_to_lds_b128(
//       int __attribute__((vector_size(16))) addrspace(1)* gaddr,
//       int __attribute__((vector_size(16))) addrspace(3)* ldsaddr,
//       imm int offset, imm int cpol)
// ---------------------------------------------------------------------------
#if defined(__AMDGCN__) && __has_builtin(__builtin_amdgcn_global_load_async_to_lds_b128) && __has_builtin(__builtin_amdgcn_s_wait_asynccnt)
#define USE_ASYNC_COPY 1
#else
#define USE_ASYNC_COPY 0
#endif

typedef int v4i_vs __attribute__((vector_size(16)));

__device__ inline void copy16_to_lds(const float* g, float* s) {
#if USE_ASYNC_COPY
    typedef __attribute__((address_space(1))) v4i_vs* as1_v4p;
    typedef __attribute__((address_space(3))) v4i_vs* as3_v4p;
    // flat LDS address: low 32 bits are the LDS byte offset (ISA 10.2 aperture rules)
    __builtin_amdgcn_global_load_async_to_lds_b128(
        (as1_v4p)(unsigned long long)(uintptr_t)g,
        (as3_v4p)(unsigned int)(uintptr_t)s,
        0, 0);
#else
    float4 v = *(const float4*)g;
    *(float4*)s = v;               // LDA=36 keeps every row 16B-aligned
#endif
}

__device__ inline void async_wait_all() {
#if USE_ASYNC_COPY
    __builtin_amdgcn_s_wait_asynccnt(0);
#endif
}

// ---------------------------------------------------------------------------
// utility: block reduction (256 threads)
// ---------------------------------------------------------------------------
__device__ inline float block_reduce_sum256(float v, float* red) {
    int t = threadIdx.x;
    red[t] = v; __syncthreads();
    for (int o = 128; o > 0; o >>= 1) { if (t < o) red[t] += red[t + o]; __syncthreads(); }
    float r = red[0]; __syncthreads();
    return r;
}

// ---------------------------------------------------------------------------
// gather 256 permuted rows of one frame and compute row squared norms
// grid: B_*nrows blocks, 256 threads
// ---------------------------------------------------------------------------
__global__ void gather_rows_norm(const float* __restrict__ src, const int* __restrict__ perm,
                                 float* __restrict__ dst, float* __restrict__ norm2,
                                 int frame, int nframes, int nrows)
{
    int blk = blockIdx.x;
    int b = blk / nrows;
    int i = blk - b * nrows;
    int p = perm[i];
    const float4* srow = (const float4*)(src + (((long)b * nframes + frame) * P_ + p) * (long)D_);
    float4* drow = (float4*)(dst + ((long)b * nrows + i) * (long)D_);
    int t = threadIdx.x;
    float4 v = srow[t];
    drow[t] = v;
    float ss = v.x * v.x + v.y * v.y + v.z * v.z + v.w * v.w;
    __shared__ float red[256];
    float total = block_reduce_sum256(ss, red);
    if (t == 0) norm2[b * nrows + i] = total;
}

// ---------------------------------------------------------------------------
// inverse clamped norms of all extra_t rows (frames 1,3,5,7)
// grid: B_*E_ blocks, 256 threads
// ---------------------------------------------------------------------------
__global__ void extra_inv_norms(const float* __restrict__ teacher, float* __restrict__ invn)
{
    int blk = blockIdx.x;
    int b = blk / E_;
    int e = blk - b * E_;
    int frame = 1 + 2 * (e >> 10);
    int p = e & (P_ - 1);
    const float4* row = (const float4*)(teacher + (((long)b * 8 + frame) * P_ + p) * (long)D_);
    int t = threadIdx.x;
    float4 v = row[t];
    float ss = v.x * v.x + v.y * v.y + v.z * v.z + v.w * v.w;
    __shared__ float red[256];
    float total = block_reduce_sum256(ss, red);
    if (t == 0) {
        float n = fmaxf(sqrtf(total), 1e-12f);
        invn[(long)b * E_ + e] = 1.0f / n;
    }
}

// ---------------------------------------------------------------------------
// fp32 WMMA GEMM:  C[b][m][n] = sum_d A[b][m][d] * B[b][n][d]   (inner dim D_=1024)
// optional per-column scale (colScale) applied at store.
// B rows may have an extra stride per group of 1024 rows (frame-interleaved extra_t).
// Block: 256 threads (8 waves), tile 64(M) x 128(N), BK=32, double-buffered LDS
// with async global->LDS copies overlapping the WMMA stream.
// Each wave: 32x32 via 2x2 v_wmma_f32_16x16x4_f32 accumulators.
// ---------------------------------------------------------------------------
#define BM 64
#define BN 128
#define BK 32
#define LDA 36   // 144B rows: 16B-aligned (async b128 legal) and bank-conflict-free reads

__global__ __launch_bounds__(256)
void wmma_gemm_nt(const float* __restrict__ A, const float* __restrict__ Bm,
                  float* __restrict__ C, const float* __restrict__ colScale,
                  int N,
                  long aBatch, long bBatch, long cBatch, long sBatch,
                  long bGroupStride)
{
    __shared__ float As[2][BM * LDA];
    __shared__ float Bs[2][BN * LDA];

    int bz    = blockIdx.z;
    int tileM = blockIdx.y * BM;
    int tileN = blockIdx.x * BN;

    const float* Ab = A  + (long)bz * aBatch;
    const float* Bb = Bm + (long)bz * bBatch + (long)(tileN >> 10) * bGroupStride;
    float*       Cb = C  + (long)bz * cBatch;

    int t    = threadIdx.x;
    int lane = t & 31;
    int wave = t >> 5;
    int wm = (wave >> 2) * 32;        // 0 / 32
    int wn = (wave & 3) * 32;         // 0 / 32 / 64 / 96
    int laneLo = lane & 15;
    int kHi    = (lane >> 4) * 2;     // lanes 0-15 -> K 0,1 ; lanes 16-31 -> K 2,3

    v8f c00 = {}, c01 = {}, c10 = {}, c11 = {};

    int lr  = t >> 3;                 // 0..31
    int lc4 = (t & 7) * 4;            // 0,4,..,28

    auto load_tile = [&](int k0, int buf) {
        #pragma unroll
        for (int rep = 0; rep < 2; ++rep) {
            int row = lr + rep * 32;
            copy16_to_lds(Ab + (long)(tileM + row) * D_ + k0 + lc4,
                          &As[buf][row * LDA + lc4]);
        }
        #pragma unroll
        for (int rep = 0; rep < 4; ++rep) {
            int row = lr + rep * 32;
            copy16_to_lds(Bb + (long)(tileN + row) * D_ + k0 + lc4,
                          &Bs[buf][row * LDA + lc4]);
        }
    };

    load_tile(0, 0);
    async_wait_all();
    __syncthreads();

    for (int k0 = 0; k0 < D_; k0 += BK) {
        int buf = (k0 / BK) & 1;
        if (k0 + BK < D_) {
            // prefetch 2 tiles ahead into L2, issue next tile into the other buffer
            __builtin_prefetch(Ab + (long)(tileM + lr) * D_ + k0 + 2 * BK + lc4, 0, 0);
            __builtin_prefetch(Bb + (long)(tileN + lr) * D_ + k0 + 2 * BK + lc4, 0, 0);
            load_tile(k0 + BK, buf ^ 1);
        }

        const float* Ap = As[buf];
        const float* Bp = Bs[buf];
        #pragma unroll
        for (int kk = 0; kk < BK; kk += 4) {
            int ks = kk + kHi;
            v2f a0, a1, b0, b1;
            a0.x = Ap[(wm + laneLo) * LDA + ks];       a0.y = Ap[(wm + laneLo) * LDA + ks + 1];
            a1.x = Ap[(wm + 16 + laneLo) * LDA + ks];  a1.y = Ap[(wm + 16 + laneLo) * LDA + ks + 1];
            b0.x = Bp[(wn + laneLo) * LDA + ks];       b0.y = Bp[(wn + laneLo) * LDA + ks + 1];
            b1.x = Bp[(wn + 16 + laneLo) * LDA + ks];  b1.y = Bp[(wn + 16 + laneLo) * LDA + ks + 1];
            c00 = __builtin_amdgcn_wmma_f32_16x16x4_f32(false, a0, false, b0, (short)0, c00, false, false);
            c01 = __builtin_amdgcn_wmma_f32_16x16x4_f32(false, a0, false, b1, (short)0, c01, false, false);
            c10 = __builtin_amdgcn_wmma_f32_16x16x4_f32(false, a1, false, b0, (short)0, c10, false, false);
            c11 = __builtin_amdgcn_wmma_f32_16x16x4_f32(false, a1, false, b1, (short)0, c11, false, false);
        }

        async_wait_all();   // my async transfers into buf^1 are complete
        __syncthreads();    // everyone done reading buf / writing buf^1
    }

    // store: C 16x16 layout -> VGPR vi: lanes 0-15 M=vi, lanes 16-31 M=vi+8; N=lane&15
    const float* scl = colScale ? (colScale + (long)bz * sBatch) : nullptr;
    int rlo = (lane >> 4) * 8;
    v8f acc[2][2] = { { c00, c01 }, { c10, c11 } };
    #pragma unroll
    for (int i = 0; i < 2; ++i) {
        #pragma unroll
        for (int j = 0; j < 2; ++j) {
            #pragma unroll
            for (int vi = 0; vi < 8; ++vi) {
                int row = tileM + wm + 16 * i + rlo + vi;
                int col = tileN + wn + 16 * j + laneLo;
                float val = acc[i][j][vi];
                if (scl) val *= scl[col];
                Cb[(long)row * N + col] = val;
            }
        }
    }
}

// ---------------------------------------------------------------------------
// top-4 per sim row. grid: B_*R_ blocks, 256 threads
// ---------------------------------------------------------------------------
__global__ void topk4_kernel(const float* __restrict__ sim, int* __restrict__ topk)
{
    int blk = blockIdx.x;
    const float* row = sim + (long)blk * E_;
    int t = threadIdx.x;

    float v[4] = { -1e30f, -1e30f, -1e30f, -1e30f };
    int   id[4] = { -1, -1, -1, -1 };
    for (int e = t; e < E_; e += 256) {
        float x = row[e];
        if (x > v[3]) {
            v[3] = x; id[3] = e;
            #pragma unroll
            for (int p = 3; p > 0; --p) {
                if (v[p] > v[p - 1]) {
                    float tv = v[p]; v[p] = v[p - 1]; v[p - 1] = tv;
                    int   ti = id[p]; id[p] = id[p - 1]; id[p - 1] = ti;
                }
            }
        }
    }

    __shared__ float sv[256 * 4];
    __shared__ int   sid[256 * 4];
    #pragma unroll
    for (int m = 0; m < 4; ++m) { sv[t * 4 + m] = v[m]; sid[t * 4 + m] = id[m]; }
    __syncthreads();

    for (int o = 128; o > 0; o >>= 1) {
        if (t < o) {
            float av[4], bv[4], ov[4]; int ai[4], bi[4], oi[4];
            #pragma unroll
            for (int m = 0; m < 4; ++m) {
                av[m] = sv[t * 4 + m];        ai[m] = sid[t * 4 + m];
                bv[m] = sv[(t + o) * 4 + m];  bi[m] = sid[(t + o) * 4 + m];
            }
            int x = 0, y = 0;
            #pragma unroll
            for (int m = 0; m < 4; ++m) {
                if (av[x] >= bv[y]) { ov[m] = av[x]; oi[m] = ai[x]; ++x; }
                else                { ov[m] = bv[y]; oi[m] = bi[y]; ++y; }
            }
            #pragma unroll
            for (int m = 0; m < 4; ++m) { sv[t * 4 + m] = ov[m]; sid[t * 4 + m] = oi[m]; }
        }
        __syncthreads();
    }
    if (t < K_) topk[blk * K_ + t] = sid[t];
}

// ---------------------------------------------------------------------------
// gather sim_high rows and compute hh, rh_t, rh_s. grid: B_*R_*K_ blocks, 256 threads
// ---------------------------------------------------------------------------
__global__ void gather_simhigh(const float* __restrict__ teacher, const int* __restrict__ topk,
                               const float* __restrict__ refT, const float* __restrict__ refS,
                               float* __restrict__ simHigh,
                               float* __restrict__ hh, float* __restrict__ rhT, float* __restrict__ rhS)
{
    int blk = blockIdx.x;                       // = (b*R_ + r)*K_ + k
    int b = blk / (R_ * K_);
    int rk = blk - b * (R_ * K_);
    int r = rk / K_;
    int e = topk[blk];
    int frame = 1 + 2 * (e >> 10);
    int p = e & (P_ - 1);

    const float4* src = (const float4*)(teacher + (((long)b * 8 + frame) * P_ + p) * (long)D_);
    const float4* rt  = (const float4*)(refT + ((long)b * R_ + r) * (long)D_);
    const float4* rs  = (const float4*)(refS + ((long)b * R_ + r) * (long)D_);
    float4* dst = (float4*)(simHigh + (long)blk * D_);

    int t = threadIdx.x;
    float4 ev = src[t]; float4 tv = rt[t]; float4 sv = rs[t];
    dst[t] = ev;
    float s_hh = ev.x * ev.x + ev.y * ev.y + ev.z * ev.z + ev.w * ev.w;
    float s_rt = ev.x * tv.x + ev.y * tv.y + ev.z * tv.z + ev.w * tv.w;
    float s_rs = ev.x * sv.x + ev.y * sv.y + ev.z * sv.z + ev.w * sv.w;

    __shared__ float red[256];
    float a = block_reduce_sum256(s_hh, red);
    float bq = block_reduce_sum256(s_rt, red);
    float c = block_reduce_sum256(s_rs, red);
    if (t == 0) { hh[blk] = a; rhT[blk] = bq; rhS[blk] = c; }
}

// ---------------------------------------------------------------------------
// RKD angle cosines + |teacher - student| accumulation (one layer)
// grid: B_*R_ blocks, 256 threads (one s per thread, k unrolled)
// writes deterministic per-block partial sums.
// ---------------------------------------------------------------------------
__device__ inline void angles3(float rr, float ss, float hh, float sr, float rh, float sh,
                               float& a1, float& a2, float& a3)
{
    float n_sr = sqrtf(fmaxf(ss - 2.f * sr + rr, 0.f));
    float n_hr = sqrtf(fmaxf(hh - 2.f * rh + rr, 0.f));
    float n_sh = sqrtf(fmaxf(ss - 2.f * sh + hh, 0.f));
    n_sr = fmaxf(n_sr, 1e-8f); n_hr = fmaxf(n_hr, 1e-8f); n_sh = fmaxf(n_sh, 1e-8f);
    a1 = (sh - rh - sr + rr) / (n_sr * n_hr);
    a2 = (sr - sh - rh + hh) / (n_hr * n_sh);
    a3 = (rh - sr - sh + ss) / (n_sr * n_sh);
}

__global__ void angle_accum(const float* __restrict__ rrT, const float* __restrict__ rrS,
                            const float* __restrict__ ssT, const float* __restrict__ ssS,
                            const float* __restrict__ hh,  const float* __restrict__ rhT,
                            const float* __restrict__ rhS,
                            const float* __restrict__ srT, const float* __restrict__ srS,
                            const float* __restrict__ shmT, const float* __restrict__ shmS,
                            float* __restrict__ partial)
{
    int blk = blockIdx.x;                  // b*R_ + r
    int b = blk / R_;
    int r = blk - b * R_;
    int s = threadIdx.x;

    float rrt = rrT[blk], rrs = rrS[blk];
    float sst = ssT[b * S_ + s], sss = ssS[b * S_ + s];
    float srt = srT[((long)b * R_ + r) * S_ + s];
    float srs = srS[((long)b * R_ + r) * S_ + s];

    float acc = 0.f;
    #pragma unroll
    for (int k = 0; k < K_; ++k) {
        float hhv = hh[blk * K_ + k];
        float rht = rhT[blk * K_ + k];
        float rhs = rhS[blk * K_ + k];
        long shIdx = ((long)b * S_ + s) * RK_ + (r * K_ + k);
        float sht = shmT[shIdx];
        float shs = shmS[shIdx];
        float a1t, a2t, a3t, a1s, a2s, a3s;
        angles3(rrt, sst, hhv, srt, rht, sht, a1t, a2t, a3t);
        angles3(rrs, sss, hhv, srs, rhs, shs, a1s, a2s, a3s);
        acc += fabsf(a1s - a1t) + fabsf(a2s - a2t) + fabsf(a3s - a3t);
    }

    __shared__ float red[256];
    float total = block_reduce_sum256(acc, red);
    if (s == 0) partial[blk] = total;
}

// final deterministic reduction: 3 layers x B_*R_ partials -> scalar loss
__global__ void final_reduce(const float* __restrict__ partial, float* __restrict__ out)
{
    __shared__ float red[256];
    int t = threadIdx.x;
    float s = 0.f;
    for (int i = t; i < 3 * B_ * R_; i += 256) s += partial[i];
    float total = block_reduce_sum256(s, red);
    if (t == 0) out[0] = total * (1.0f / 3145728.0f);   // / (3*B*R*S*K)
}

// ---------------------------------------------------------------------------
extern "C" void kernel_launch(void* const* d_in, const int* in_sizes, int n_in,
                              void* d_out, int out_size, void* d_ws, size_t ws_size,
                              hipStream_t stream)
{
    const float* teacher   = (const float*)d_in[0];
    const float* student   = (const float*)d_in[1];
    const int*   refPerm   = (const int*)d_in[2];
    const int*   sharedPerm= (const int*)d_in[3];
    float* out = (float*)d_out;

    char* ws = (char*)d_ws;
    size_t off = 0;
    auto alloc = [&](size_t bytes) -> void* {
        void* p = ws + off;
        off += (bytes + 255) & ~(size_t)255;
        return p;
    };
    float* refT    = (float*)alloc(sizeof(float) * (size_t)B_ * R_ * D_);
    float* refS    = (float*)alloc(sizeof(float) * (size_t)B_ * R_ * D_);
    float* rrT     = (float*)alloc(sizeof(float) * B_ * R_);
    float* rrS     = (float*)alloc(sizeof(float) * B_ * R_);
    float* invNorm = (float*)alloc(sizeof(float) * (size_t)B_ * E_);
    float* sim     = (float*)alloc(sizeof(float) * (size_t)B_ * R_ * E_);
    int*   topk    = (int*)  alloc(sizeof(int)   * (size_t)B_ * R_ * K_);
    float* simHigh = (float*)alloc(sizeof(float) * (size_t)B_ * R_ * K_ * D_);
    float* hh      = (float*)alloc(sizeof(float) * B_ * R_ * K_);
    float* rhT     = (float*)alloc(sizeof(float) * B_ * R_ * K_);
    float* rhS     = (float*)alloc(sizeof(float) * B_ * R_ * K_);
    float* shT     = (float*)alloc(sizeof(float) * (size_t)B_ * S_ * D_);
    float* shS     = (float*)alloc(sizeof(float) * (size_t)B_ * S_ * D_);
    float* ssT     = (float*)alloc(sizeof(float) * B_ * S_);
    float* ssS     = (float*)alloc(sizeof(float) * B_ * S_);
    float* srT     = (float*)alloc(sizeof(float) * (size_t)B_ * R_ * S_);
    float* srS     = (float*)alloc(sizeof(float) * (size_t)B_ * R_ * S_);
    float* shmT    = (float*)alloc(sizeof(float) * (size_t)B_ * S_ * RK_);
    float* shmS    = (float*)alloc(sizeof(float) * (size_t)B_ * S_ * RK_);
    float* partial = (float*)alloc(sizeof(float) * 3 * B_ * R_);
    (void)ws_size; (void)in_sizes; (void)n_in; (void)out_size;

    // 1. gathers + norms
    gather_rows_norm<<<B_ * R_, 256, 0, stream>>>(teacher, refPerm, refT, rrT, 0, 8, R_);
    gather_rows_norm<<<B_ * R_, 256, 0, stream>>>(student, refPerm, refS, rrS, 0, 4, R_);
    extra_inv_norms<<<B_ * E_, 256, 0, stream>>>(teacher, invNorm);

    // 2. sim = refT . extraT^T, scaled by 1/|e| (ranking-equivalent to l2n.l2n)
    //    extra row e lives at teacher + b*8PD + PD + (e>>10)*PD + e*D
    {
        dim3 g(E_ / BN, R_ / BM, B_);
        wmma_gemm_nt<<<g, 256, 0, stream>>>(refT, teacher + (size_t)P_ * D_, sim, invNorm,
                                            E_,
                                            (long)R_ * D_, (long)8 * P_ * D_,
                                            (long)R_ * E_, (long)E_,
                                            (long)P_ * D_);
    }

    // 3. top-4 and sim_high gather (+hh, rh)
    topk4_kernel<<<B_ * R_, 256, 0, stream>>>(sim, topk);
    gather_simhigh<<<B_ * R_ * K_, 256, 0, stream>>>(teacher, topk, refT, refS, simHigh, hh, rhT, rhS);

    // 4. three shared layers
    const int teacherFrames[3] = { 2, 4, 6 };
    const int studentFrames[3] = { 1, 2, 3 };
    for (int l = 0; l < 3; ++l) {
        gather_rows_norm<<<B_ * S_, 256, 0, stream>>>(teacher, sharedPerm, shT, ssT, teacherFrames[l], 8, S_);
        gather_rows_norm<<<B_ * S_, 256, 0, stream>>>(student, sharedPerm, shS, ssS, studentFrames[l], 4, S_);

        dim3 g2(S_ / BN, R_ / BM, B_);    // sr = ref . shared^T  [R x S]
        wmma_gemm_nt<<<g2, 256, 0, stream>>>(refT, shT, srT, nullptr, S_,
                                             (long)R_ * D_, (long)S_ * D_, (long)R_ * S_, 0, 0);
        wmma_gemm_nt<<<g2, 256, 0, stream>>>(refS, shS, srS, nullptr, S_,
                                             (long)R_ * D_, (long)S_ * D_, (long)R_ * S_, 0, 0);

        dim3 g3(RK_ / BN, S_ / BM, B_);   // shm = shared . simHigh^T  [S x R*K]
        wmma_gemm_nt<<<g3, 256, 0, stream>>>(shT, simHigh, shmT, nullptr, RK_,
                                             (long)S_ * D_, (long)RK_ * D_, (long)S_ * RK_, 0, 0);
        wmma_gemm_nt<<<g3, 256, 0, stream>>>(shS, simHigh, shmS, nullptr, RK_,
                                             (long)S_ * D_, (long)RK_ * D_, (long)S_ * RK_, 0, 0);

        angle_accum<<<B_ * R_, 256, 0, stream>>>(rrT, rrS, ssT, ssS, hh, rhT, rhS,
                                                 srT, srS, shmT, shmS,
                                                 partial + (size_t)l * B_ * R_);
    }

    // 5. deterministic final reduction
    final_reduce<<<1, 256, 0, stream>>>(partial, out);
}